// LT_unit_1640677507419
// MI455X (gfx1250) — compile-verified
//
#include <hip/hip_runtime.h>
#include <stdint.h>

// ---------------------------------------------------------------------------
// Conv3x3(8->8, pad=1) + BatchNorm(train) + ReLU on MI455X (gfx1250, wave32)
//  * Conv via V_WMMA_F32_16X16X4_F32 (fp32-exact): M=16 out-ch (8 real),
//    N=16 pixels, K=72 chained as 18 K=4 steps; 4 output rows per block.
//  * x slab staged to LDS with CDNA5 GLOBAL_LOAD_ASYNC_TO_LDS_B32 (ASYNCcnt).
//  * BN bias cancels analytically -> lin_b unused.
//  * Deterministic stats: register accum -> wave butterfly -> block partials
//    -> single-block fixed-order tree reduce -> in-place scale/shift + ReLU.
// ---------------------------------------------------------------------------

typedef __attribute__((ext_vector_type(2))) float v2f;
typedef __attribute__((ext_vector_type(8))) float v8f;

#define BN_    16
#define CIN_   8
#define COUT_  8
#define HH_    512
#define WW_    512
#define TILEW_ 128                           // pixels per block row-tile
#define ROWS_  4                             // output rows per block
#define NBLK_  (BN_ * (HH_ / ROWS_) * (WW_ / TILEW_))   // 8192
#define XS_W   130                           // 128 + 2 halo cols
#define XS_R   (ROWS_ + 2)                   // 6 slab rows
#define XS_N   (CIN_ * XS_R * XS_W)          // 6240 floats (~25 KB)
#define WF_N   (18 * 2 * 32)                 // 1152 floats

// LDS float offset of x element for output row rt, reduction idx r=c*9+kh*3+kw
static __device__ __forceinline__ int xoff(int rt, int r) {
    int c  = r / 9;
    int kk = r - c * 9;
    int kh = kk / 3;
    int kw = kk - kh * 3;
    return c * (XS_R * XS_W) + (rt + kh) * XS_W + kw;
}

__global__ __launch_bounds__(256) void conv_wmma_kernel(
    const float* __restrict__ x, const float* __restrict__ lw,
    float* __restrict__ y, float* __restrict__ pS, float* __restrict__ pQ)
{
    __shared__ float xs[XS_N];      // x slab: [c][6 rows][130 cols]
    __shared__ float wf[WF_N];      // A fragments: [step][lane][2] (b64/lane)
    __shared__ float bsum[8][8];    // [wave][channel]
    __shared__ float bsq[8][8];

    const int w0  = blockIdx.x * TILEW_;
    const int h0  = blockIdx.y * ROWS_;
    const int b   = blockIdx.z;
    const int tid = threadIdx.x;

    // ---- stage x slab via async global->LDS; zero-fill halo OOB ----
    for (int e = tid; e < XS_N; e += 256) {
        int c   = e / (XS_R * XS_W);
        int rem = e - c * (XS_R * XS_W);
        int j   = rem / XS_W;
        int wi  = rem - j * XS_W;
        int hin = h0 - 1 + j;
        int win = w0 - 1 + wi;
        if ((unsigned)hin < (unsigned)HH_ && (unsigned)win < (unsigned)WW_) {
            unsigned lds  = (unsigned)(uintptr_t)&xs[e];   // low 32b = LDS addr
            unsigned goff = (unsigned)((((b * CIN_ + c) * HH_ + hin) * WW_ + win) * 4);
            asm volatile("global_load_async_to_lds_b32 %0, %1, %2"
                         :: "v"(lds), "v"(goff), "s"(x) : "memory");
        } else {
            xs[e] = 0.0f;
        }
    }

    // ---- stage weight A-fragments pre-swizzled, lane-major pairs ----
    // A 16x4 f32: lane(l<16)=row M=l; VGPR v + half hf -> K = 2*hf + v
    for (int e = tid; e < WF_N; e += 256) {
        int step = e >> 6;
        int le   = (e >> 1) & 31;
        int vv   = e & 1;
        int hfe  = le >> 4;
        int m    = le & 15;                  // out channel; rows 8..15 -> 0
        int r    = step * 4 + 2 * hfe + vv;
        float wv = 0.0f;
        if (m < COUT_) {
            int c  = r / 9;
            int kk = r - c * 9;
            wv = lw[(c * COUT_ + m) * 9 + kk];     // lin_w[c][m][kk]
        }
        wf[e] = wv;
    }

    asm volatile("s_wait_asynccnt 0" ::: "memory");
    __syncthreads();

    const int lane = tid & 31;
    const int wave = tid >> 5;
    const int hf   = lane >> 4;
    const int nn   = lane & 15;
    const int px   = wave * 16 + nn;         // pixel within 128-wide tile

    float s_[COUT_], q_[COUT_];
#pragma unroll
    for (int i = 0; i < COUT_; ++i) { s_[i] = 0.0f; q_[i] = 0.0f; }

#pragma unroll
    for (int rt = 0; rt < ROWS_; ++rt) {
        v8f acc = {0.f, 0.f, 0.f, 0.f, 0.f, 0.f, 0.f, 0.f};
#pragma unroll
        for (int step = 0; step < 18; ++step) {
            v2f a = *(const v2f*)&wf[step * 64 + lane * 2];   // one ds_load_b64
            int r0 = step * 4;
            int o0 = hf ? xoff(rt, r0 + 2) : xoff(rt, r0);    // folds to imm
            int o1 = hf ? xoff(rt, r0 + 3) : xoff(rt, r0 + 1);
            v2f bm;
            bm.x = xs[o0 + px];
            bm.y = xs[o1 + px];
            acc = __builtin_amdgcn_wmma_f32_16x16x4_f32(
                false, a, false, bm, (short)0, acc, false, false);
        }

        // store: VGPR i = channel i for lanes 0..15 (pixel = lane), coalesced
        if (lane < 16) {
            int col = w0 + wave * 16 + lane;
            int h   = h0 + rt;
#pragma unroll
            for (int i = 0; i < COUT_; ++i)
                y[((b * COUT_ + i) * HH_ + h) * WW_ + col] = acc[i];
        }
#pragma unroll
        for (int i = 0; i < COUT_; ++i) {
            float v = (lane < 16) ? acc[i] : 0.0f;   // upper half is exact 0
            s_[i] += v;
            q_[i] += v * v;
        }
    }

    // ---- deterministic wave butterfly + fixed-order cross-wave partials ----
#pragma unroll
    for (int i = 0; i < COUT_; ++i) {
        float v = s_[i], q = q_[i];
#pragma unroll
        for (int off = 16; off >= 1; off >>= 1) {
            v += __shfl_xor(v, off, 32);
            q += __shfl_xor(q, off, 32);
        }
        if (lane == 0) { bsum[wave][i] = v; bsq[wave][i] = q; }
    }
    __syncthreads();

    const int bid = (blockIdx.z * gridDim.y + blockIdx.y) * gridDim.x + blockIdx.x;
    if (tid < 8) {
        float s = 0.0f;
#pragma unroll
        for (int w = 0; w < 8; ++w) s += bsum[w][tid];
        pS[bid * 8 + tid] = s;
    } else if (tid < 16) {
        int ch = tid - 8;
        float s = 0.0f;
#pragma unroll
        for (int w = 0; w < 8; ++w) s += bsq[w][ch];
        pQ[bid * 8 + ch] = s;
    }
}

__global__ __launch_bounds__(256) void bn_finalize_kernel(
    const float* __restrict__ pS, const float* __restrict__ pQ,
    const float* __restrict__ gamma, const float* __restrict__ beta,
    float* __restrict__ ss, int P)
{
    __shared__ float r0[256];
    __shared__ float r1[256];
    const int t = threadIdx.x;
    for (int ch = 0; ch < COUT_; ++ch) {
        float s = 0.0f, q = 0.0f;
        for (int j = t; j < P; j += 256) {       // fixed per-thread order
            s += pS[j * 8 + ch];
            q += pQ[j * 8 + ch];
        }
        r0[t] = s; r1[t] = q;
        __syncthreads();
        for (int st = 128; st > 0; st >>= 1) {   // fixed tree order
            if (t < st) { r0[t] += r0[t + st]; r1[t] += r1[t + st]; }
            __syncthreads();
        }
        if (t == 0) {
            const float N = (float)BN_ * (float)HH_ * (float)WW_;
            float mean = r0[0] / N;
            float var  = r1[0] / N - mean * mean;
            float rstd = rsqrtf(var + 1e-5f);
            float sc   = gamma[ch] * rstd;
            ss[ch]     = sc;
            ss[8 + ch] = beta[ch] - mean * sc;
        }
        __syncthreads();
    }
}

__global__ __launch_bounds__(256) void bn_apply_kernel(
    float* __restrict__ y, const float* __restrict__ ss)
{
    const int idx4 = blockIdx.x * 256 + threadIdx.x;   // float4 index
    const int ch   = (idx4 >> 16) & 7;                 // (idx4*4)/(H*W) % 8
    const float sc = ss[ch];
    const float sh = ss[8 + ch];
    float4 v = ((const float4*)y)[idx4];
    v.x = fmaxf(v.x * sc + sh, 0.0f);
    v.y = fmaxf(v.y * sc + sh, 0.0f);
    v.z = fmaxf(v.z * sc + sh, 0.0f);
    v.w = fmaxf(v.w * sc + sh, 0.0f);
    ((float4*)y)[idx4] = v;
}

extern "C" void kernel_launch(void* const* d_in, const int* in_sizes, int n_in,
                              void* d_out, int out_size, void* d_ws, size_t ws_size,
                              hipStream_t stream)
{
    const float* x     = (const float*)d_in[0];
    const float* lw    = (const float*)d_in[1];
    // d_in[2] (lin_b): constant per-channel conv bias cancels in BatchNorm
    const float* gamma = (const float*)d_in[3];
    const float* beta  = (const float*)d_in[4];

    float* y  = (float*)d_out;
    float* pS = (float*)d_ws;                  // NBLK_*8 floats
    float* pQ = pS + (size_t)NBLK_ * 8;        // NBLK_*8 floats
    float* ss = pQ + (size_t)NBLK_ * 8;        // scale[8], shift[8]

    conv_wmma_kernel<<<dim3(WW_ / TILEW_, HH_ / ROWS_, BN_), 256, 0, stream>>>(
        x, lw, y, pS, pQ);
    bn_finalize_kernel<<<1, 256, 0, stream>>>(pS, pQ, gamma, beta, ss, NBLK_);
    bn_apply_kernel<<<(BN_ * COUT_ * HH_ * WW_) / (256 * 4), 256, 0, stream>>>(y, ss);
}